// PointNet2GroupingLayer_53506702574030
// MI455X (gfx1250) — compile-verified
//
#include <hip/hip_runtime.h>

// ---------------------------------------------------------------------------
// PointNet++ grouping on MI455X (gfx1250, wave32).
//
// Ball query distances via V_WMMA_F32_16X16X4_F32:
//   d2[m][n] = |c_m|^2 - 2 c_m . p_n + |p_n|^2
//   A (16x4, M=centers):  row m = (-2cx, -2cy, -2cz, 1)
//   B (4x16, N=points):   col n = ( px,   py,   pz,  |p|^2)
//   C (16x16):            C[m][n] = |c_m|^2
// One WMMA = a full 16x16 tile of exact squared distances.
//
// VGPR layouts (ISA 7.12.2, 32-bit, wave32):
//   A 16x4 : lanes 0-15 -> M=lane,   v0=A[m][0], v1=A[m][1]
//            lanes16-31 -> M=lane-16,v0=A[m][2], v1=A[m][3]
//   B 4x16 : lanes 0-15 -> N=lane,   v0=B[0][n], v1=B[1][n]
//            lanes16-31 -> N=lane-16,v0=B[2][n], v1=B[3][n]
//   C/D    : VGPR r: lanes 0-15 -> (M=r,   N=lane)
//                    lanes16-31 -> (M=r+8, N=lane-16)
//
// Roofline: output stream = 4*2048*67*64*4B ~= 140 MB -> ~6 us floor at
// 23.3 TB/s; gather tables (16 MB features + raw (B,N,3) xyz) live in the 192 MB
// L2.  Ball query scans ~2-3K of 16384 points per center before the uniform
// ballot early-exit fires, so the WMMA phase is far off the critical path.
// ---------------------------------------------------------------------------

typedef __attribute__((ext_vector_type(2))) float v2f;
typedef __attribute__((ext_vector_type(4))) float v4f;
typedef __attribute__((ext_vector_type(8))) float v8f;
typedef __attribute__((ext_vector_type(4))) int   v4i;

constexpr int   kB       = 4;
constexpr int   kN       = 16384;
constexpr int   kNPoint  = 2048;
constexpr int   kC       = 64;
constexpr int   kNSample = 64;
constexpr float kR2      = 0.2f * 0.2f;
constexpr int   kRows    = 3 + kC;          // 67 output channels
constexpr int   kWaves   = 4;               // waves per ball-query block

// --------------------------------------------------------------------------
// Kernel 0: pack xyz into (x, y, z, |p|^2) float4 so each WMMA B-fragment is
// a single global_load_b128 (both half-waves broadcast the same 256B tile).
// --------------------------------------------------------------------------
__global__ void pack_points(const float* __restrict__ xyz, v4f* __restrict__ w) {
  int i = blockIdx.x * blockDim.x + threadIdx.x;   // over kB * kN points
  if (i < kB * kN) {
    float x = xyz[i * 3 + 0];
    float y = xyz[i * 3 + 1];
    float z = xyz[i * 3 + 2];
    v4f p; p.x = x; p.y = y; p.z = z; p.w = x * x + y * y + z * z;
    w[i] = p;
  }
}

// --------------------------------------------------------------------------
// Kernel 1: ball query. One wave32 handles 16 centers; scans point tiles of
// 16 in index order via WMMA, ballots the in-ball predicate per accumulator
// VGPR, and lanes 0-15 (one per center row) append indices LSB-first until
// 64 are collected.  Index lists build in LDS (row stride 68 ints -> append
// addresses hit distinct banks) and flush with coalesced 128-bit stores.
// --------------------------------------------------------------------------
__global__ void ball_query_wmma(const float* __restrict__ new_xyz,  // (B,NP,3)
                                const v4f*  __restrict__ w,         // (B*N)
                                int*        __restrict__ idx)       // (B,NP,64)
{
  __shared__ int sbuf[kWaves][16][68];   // 64 slots + pad (bank-conflict-free)

  const int wv    = threadIdx.x >> 5;
  const int group = blockIdx.x * kWaves + wv;
  const int lane  = threadIdx.x & 31;
  const int b     = group / (kNPoint / 16);
  const int g     = group % (kNPoint / 16);
  const int centerBase = g * 16;
  const int half  = lane >> 4;          // 0: lanes 0-15, 1: lanes 16-31
  const int l16   = lane & 15;

  // ---- A fragment: my center (index centerBase + l16) ----
  const float* cp = new_xyz + ((size_t)b * kNPoint + centerBase + l16) * 3;
  const float cx = cp[0], cy = cp[1], cz = cp[2];
  v2f A;
  A.x = half ? (-2.0f * cz) : (-2.0f * cx);
  A.y = half ? 1.0f         : (-2.0f * cy);

  // ---- C fragment: C[m][n] = |c_m|^2 ; VGPR r holds row r + half*8 ----
  v8f Cm;
#pragma unroll
  for (int r = 0; r < 8; ++r) {
    const int m = r + half * 8;
    const float* cc = new_xyz + ((size_t)b * kNPoint + centerBase + m) * 3;
    const float x = cc[0], y = cc[1], z = cc[2];
    Cm[r] = x * x + y * y + z * z;
  }

  int  cnt   = 0;
  int  first = 0;
  bool has   = false;
  const v4f* wb = w + (size_t)b * kN;

  for (int tile = 0; tile < kN / 16; ++tile) {
    // Stream upcoming tiles through the gfx1250 prefetch path.
    __builtin_prefetch((const void*)(wb + (tile + 4) * 16), 0, 1);

    // ---- B fragment: 16 points of this tile (one b128 per lane) ----
    const v4f p = wb[tile * 16 + l16];
    v2f Bm;
    Bm.x = half ? p.z : p.x;
    Bm.y = half ? p.w : p.y;

    // EXEC is all-ones here (uniform loop, reconverged control flow).
    v8f D = __builtin_amdgcn_wmma_f32_16x16x4_f32(
        /*neg_a=*/false, A, /*neg_b=*/false, Bm,
        /*c_mod=*/(short)0, Cm, /*reuse_a=*/false, /*reuse_b=*/false);

    // Ballot per accumulator VGPR: bits[15:0] -> row r, bits[31:16] -> row r+8.
    unsigned masks[8];
#pragma unroll
    for (int r = 0; r < 8; ++r)
      masks[r] = (unsigned)__ballot(D[r] < kR2);

    // Lane l16 (lower half) owns center row l16: append in index order.
    // Row l16 lives in masks[l16 & 7]; low/high 16 bits select by l16>=8.
    if (half == 0) {
      const unsigned mm = masks[l16 & 7];
      unsigned m16 = (l16 < 8) ? (mm & 0xFFFFu) : (mm >> 16);
      while (m16 && cnt < kNSample) {
        const int bit = __builtin_ctz(m16);
        const int pi  = tile * 16 + bit;
        if (!has) { has = true; first = pi; }
        sbuf[wv][l16][cnt++] = pi;
        m16 &= m16 - 1;
      }
    }

    // Uniform early exit once every center row has 64 hits.
    const bool need = (half == 0) && (cnt < kNSample);
    if (__ballot(need) == 0) break;
  }

  if (half == 0) {
    const int fill = has ? first : 0;   // CUDA-kernel pad semantics
    for (int s = cnt; s < kNSample; ++s) sbuf[wv][l16][s] = fill;
  }

  // Flush 16 rows x 64 ints with coalesced 128-bit stores (same wave wrote
  // the LDS data, so only a dscnt wait is needed -- no barrier).
  int* gout = idx + ((size_t)b * kNPoint + centerBase) * kNSample;
  for (int t4 = lane; t4 < 256; t4 += 32) {
    const int row  = t4 >> 4;           // 0..15
    const int col4 = t4 & 15;           // quad within row
    const v4i q = *(const v4i*)&sbuf[wv][row][col4 * 4];
    *(v4i*)(gout + row * kNSample + col4 * 4) = q;
  }
}

// --------------------------------------------------------------------------
// Kernel 2: gather + emit (B, npoint, 3+C, nsample).  One block per (b, j);
// rows 0..2 are centered xyz, rows 3..66 are gathered features.  Each thread
// owns a 4-sample quad of one row per iteration -> one global_store_b128,
// 512B contiguous per wave.  This phase is a clean HBM-store-bound stream.
// --------------------------------------------------------------------------
__global__ void gather_out(const float* __restrict__ xyz,      // (B,N,3)
                           const float* __restrict__ new_xyz,  // (B,NP,3)
                           const float* __restrict__ feat,     // (B,C,N)
                           const int*   __restrict__ idx,      // (B,NP,64)
                           float*       __restrict__ out)      // (B,NP,67,64)
{
  const int bj = blockIdx.x;            // 0 .. B*NP-1
  const int b  = bj / kNPoint;
  const int q      = threadIdx.x & 15;  // sample quad: s = 4q .. 4q+3
  const int cphase = threadIdx.x >> 4;  // 0..15 row phase

  __shared__ int   sidx[kNSample];
  __shared__ float scen[3];
  if (threadIdx.x < kNSample) sidx[threadIdx.x] = idx[(size_t)bj * kNSample + threadIdx.x];
  if (threadIdx.x < 3)        scen[threadIdx.x] = new_xyz[(size_t)bj * 3 + threadIdx.x];
  __syncthreads();

  const int id0 = sidx[q * 4 + 0];
  const int id1 = sidx[q * 4 + 1];
  const int id2 = sidx[q * 4 + 2];
  const int id3 = sidx[q * 4 + 3];

  float*       outbj = out  + (size_t)bj * kRows * kNSample;
  const float* xb    = xyz  + (size_t)b * kN * 3;
  const float* fb    = feat + (size_t)b * kC * kN;

  for (int c = cphase; c < kRows; c += 16) {
    v4f v;
    if (c < 3) {
      const float cen = scen[c];
      v.x = xb[id0 * 3 + c] - cen;
      v.y = xb[id1 * 3 + c] - cen;
      v.z = xb[id2 * 3 + c] - cen;
      v.w = xb[id3 * 3 + c] - cen;
    } else {
      const float* fr = fb + (size_t)(c - 3) * kN;
      v.x = fr[id0]; v.y = fr[id1]; v.z = fr[id2]; v.w = fr[id3];
    }
    *(v4f*)(outbj + c * kNSample + q * 4) = v;   // global_store_b128
  }
}

// --------------------------------------------------------------------------
extern "C" void kernel_launch(void* const* d_in, const int* in_sizes, int n_in,
                              void* d_out, int out_size, void* d_ws, size_t ws_size,
                              hipStream_t stream) {
  const float* xyz     = (const float*)d_in[0];   // (4,16384,3)
  const float* new_xyz = (const float*)d_in[1];   // (4,2048,3)
  const float* feat    = (const float*)d_in[2];   // (4,64,16384)
  float*       out     = (float*)d_out;           // (4,2048,67,64)

  // Workspace: [0, 1MB)  packed points (B*N v4f)
  //            [1MB,3MB) idx (B*NP*64 int)
  v4f* w   = (v4f*)d_ws;
  int* idx = (int*)((char*)d_ws + (size_t)kB * kN * sizeof(v4f));

  pack_points<<<(kB * kN + 255) / 256, 256, 0, stream>>>(xyz, w);

  const int groups = kB * (kNPoint / 16);          // 512 waves
  ball_query_wmma<<<groups / kWaves, kWaves * 32, 0, stream>>>(new_xyz, w, idx);

  gather_out<<<kB * kNPoint, 256, 0, stream>>>(xyz, new_xyz, feat, idx, out);
}